// MoEElementFusion_72035191489054
// MI455X (gfx1250) — compile-verified
//
#include <hip/hip_runtime.h>
#include <hip/hip_bf16.h>
#include <math.h>

// ---------------------------------------------------------------------------
// MoE (2 views, 16 experts, top-4, D=1024, H=4096) for gfx1250 (MI455X).
// fp32 path using V_WMMA_F32_16X16X4_F32. Routed (top-k) computation:
// identical result to the dense reference, 4x fewer FLOPs.
// ---------------------------------------------------------------------------

#define Dm   1024
#define Em   16
#define Hm   4096
#define Nm   2048          // B*L tokens per view
#define GTOT (2 * Nm)      // tokens across both views
#define CAP  4096          // max tokens routed to one expert (both views)
#define HSTR 18            // padded LDS stride for hidden buffer (bank-conflict-free)

typedef __attribute__((ext_vector_type(2))) float v2f;
typedef __attribute__((ext_vector_type(8))) float v8f;

// ---------------------------------------------------------------------------
// 1) Laplace gate: logits = -(||h||^2 - 2 h.k_e + ||k_e||^2) + h.rw_e + rb_e
//    top-4 + softmax.  One block per (view,token).  Deterministic reductions.
// ---------------------------------------------------------------------------
__global__ __launch_bounds__(256) void gate_kernel(
    const float* __restrict__ v0, const float* __restrict__ v1,
    const float* __restrict__ keys,
    const float* __restrict__ rw0, const float* __restrict__ rb0,
    const float* __restrict__ rw1, const float* __restrict__ rb1,
    int* __restrict__ gidx, float* __restrict__ gw) {
  __shared__ float hs[Dm];
  __shared__ float logits[Em];

  int g    = blockIdx.x;           // global token id in [0, 2N)
  int view = g >> 11;              // g / Nm
  int t    = g & (Nm - 1);
  const float* src = (view ? v1 : v0) + (size_t)t * Dm;
  const float* rw  = view ? rw1 : rw0;
  const float* rb  = view ? rb1 : rb0;

  for (int i = threadIdx.x; i < Dm; i += blockDim.x) hs[i] = src[i];
  __syncthreads();

  int wave = threadIdx.x >> 5;
  int lane = threadIdx.x & 31;

  // 8 waves x 2 experts each
  for (int ei = 0; ei < 2; ++ei) {
    int e = wave * 2 + ei;
    const float* ke = keys + (size_t)e * Dm;
    const float* re = rw   + (size_t)e * Dm;
    float dk = 0.f, dr = 0.f, kk = 0.f, hh = 0.f;
    for (int i = lane; i < Dm; i += 32) {
      float x = hs[i], kv = ke[i];
      dk += x * kv;
      dr += x * re[i];
      kk += kv * kv;
      hh += x * x;
    }
    for (int o = 16; o > 0; o >>= 1) {
      dk += __shfl_xor(dk, o, 32);
      dr += __shfl_xor(dr, o, 32);
      kk += __shfl_xor(kk, o, 32);
      hh += __shfl_xor(hh, o, 32);
    }
    if (lane == 0) logits[e] = -(hh - 2.f * dk + kk) + dr + rb[e];
  }
  __syncthreads();

  if (threadIdx.x == 0) {
    float lv[Em];
    for (int e = 0; e < Em; ++e) lv[e] = logits[e];
    int   id[4];
    float vl[4];
    bool  used[Em] = {};
    for (int k = 0; k < 4; ++k) {            // descending, ties -> lowest idx
      int best = 0; float bv = -3.4e38f;
      for (int e = 0; e < Em; ++e)
        if (!used[e] && lv[e] > bv) { bv = lv[e]; best = e; }
      used[best] = true; id[k] = best; vl[k] = bv;
    }
    float mx = vl[0], s = 0.f, ex[4];
    for (int k = 0; k < 4; ++k) { ex[k] = expf(vl[k] - mx); s += ex[k]; }
    float inv = 1.f / s;
    for (int k = 0; k < 4; ++k) {
      gidx[g * 4 + k] = id[k];
      gw[g * 4 + k]   = ex[k] * inv;
    }
  }
}

// ---------------------------------------------------------------------------
// 2) Deterministic per-expert compaction: one wave per expert, ballot+popcount
//    keeps token order -> bitwise-stable lists with no atomics.
// ---------------------------------------------------------------------------
__global__ void compact_kernel(const int* __restrict__ gidx,
                               int* __restrict__ lists,
                               int* __restrict__ counts) {
  int e    = blockIdx.x;
  int lane = threadIdx.x;          // 32 threads
  int base = 0;
  for (int c = 0; c < GTOT; c += 32) {
    int g = c + lane;
    int4 id = ((const int4*)gidx)[g];
    int k = -1;
    if      (id.x == e) k = 0;
    else if (id.y == e) k = 1;
    else if (id.z == e) k = 2;
    else if (id.w == e) k = 3;
    unsigned int b32 = (unsigned int)__ballot(k >= 0);
    int prefix = __popc(b32 & ((1u << lane) - 1u));
    if (k >= 0) lists[e * CAP + base + prefix] = (g << 2) | k;
    base += __popc(b32);
  }
  if (lane == 0) counts[e] = base;
}

// ---------------------------------------------------------------------------
// 3) Fused expert FFN with fp32 WMMA.
//    CTA = (16 gathered tokens, expert e).  256 threads = 8 waves.
//    LDS: Xs[1024][16] (64 KB) + Hs[2048][18-padded] (144 KB).
//    FC1 in two 2048-wide halves, 4 N-tiles per A-fragment (independent WMMA
//    chains); FC2 accumulated in registers across halves, 8 N-tiles per A.
//    grid = (CAP/16, Em), expert-major so one expert's 32 MB of weights stays
//    L2-resident (192 MB L2) while its token tiles stream.
// ---------------------------------------------------------------------------
__global__ __launch_bounds__(256) void expert_kernel(
    const float* __restrict__ v0, const float* __restrict__ v1,
    const float* __restrict__ W1, const float* __restrict__ b1,
    const float* __restrict__ W2, const float* __restrict__ b2,
    const int* __restrict__ lists, const int* __restrict__ counts,
    const float* __restrict__ gw, float* __restrict__ contrib) {
  extern __shared__ float lds[];
  float* Xs = lds;                 // [k][m] : 1024 * 16
  float* Hs = lds + 1024 * 16;     // [k'][m]: 2048 * HSTR
  __shared__ int   gl_g[16];
  __shared__ int   gl_k[16];
  __shared__ float gl_w[16];

  int e     = blockIdx.y;
  int cnt   = counts[e];
  int start = blockIdx.x * 16;
  if (start >= cnt) return;
  int rows = min(16, cnt - start);

  int tid  = threadIdx.x;
  int lane = tid & 31;
  int wave = tid >> 5;
  int lhi  = (lane < 16) ? 0 : 1;  // which half of the wave
  int lnm  = lane & 15;

  if (tid < 16) {
    if (tid < rows) {
      int entry = lists[e * CAP + start + tid];
      int g = entry >> 2, k = entry & 3;
      gl_g[tid] = g; gl_k[tid] = k; gl_w[tid] = gw[g * 4 + k];
    } else {
      gl_g[tid] = -1; gl_k[tid] = 0; gl_w[tid] = 0.f;
    }
  }
  __syncthreads();

  // Gather X tile transposed into LDS: Xs[k*16 + m]
  for (int idx = tid; idx < 16 * Dm; idx += 256) {
    int m  = idx >> 10;
    int kk = idx & (Dm - 1);
    int g  = gl_g[m];
    float val = 0.f;
    if (g >= 0) {
      const float* src = (g < Nm) ? (v0 + (size_t)g * Dm)
                                  : (v1 + (size_t)(g - Nm) * Dm);
      val = src[kk];
    }
    Xs[kk * 16 + m] = val;
  }
  __syncthreads();

  const float* W1e = W1 + (size_t)e * Dm * Hm;
  const float* W2e = W2 + (size_t)e * Hm * Dm;

  v8f acc2[8];
  for (int j = 0; j < 8; ++j) acc2[j] = (v8f)(0.f);

  int moff = lhi * 8;

  for (int half = 0; half < 2; ++half) {
    // ---- FC1 + bias + exact GELU -> Hs (this half's 2048 hidden cols) ----
    // 16 N-tiles per wave, processed as 4 groups of 4 sharing each A-frag.
    for (int grp = 0; grp < 4; ++grp) {
      int nbase = (wave * 16 + grp * 4) * 16 + lnm;  // hidden col within half
      v8f acc[4];
      #pragma unroll
      for (int j = 0; j < 4; ++j) acc[j] = (v8f)(0.f);
      #pragma unroll 2
      for (int k0 = 0; k0 < Dm; k0 += 4) {
        int kb = k0 + lhi * 2;
        v2f a;
        a.x = Xs[kb * 16 + lnm];
        a.y = Xs[(kb + 1) * 16 + lnm];
        const float* wp = W1e + (size_t)kb * Hm + half * 2048 + nbase;
        #pragma unroll
        for (int j = 0; j < 4; ++j) {
          v2f b;
          b.x = wp[j * 16];
          b.y = wp[j * 16 + Hm];
          acc[j] = __builtin_amdgcn_wmma_f32_16x16x4_f32(
              false, a, false, b, (short)0, acc[j], false, false);
        }
      }
      #pragma unroll
      for (int j = 0; j < 4; ++j) {
        int nw = nbase + j * 16;               // hidden col within half
        int n  = half * 2048 + nw;             // global hidden col
        float bv = b1[(size_t)e * Hm + n];
        #pragma unroll
        for (int r = 0; r < 8; ++r) {
          float x  = acc[j][r] + bv;
          float gl = 0.5f * x * (1.0f + erff(x * 0.70710678118654752f));
          Hs[nw * HSTR + (r + moff)] = gl;
        }
      }
    }
    __syncthreads();

    // ---- FC2 partial accumulation over this half's K=2048 ----
    #pragma unroll 2
    for (int k0 = 0; k0 < 2048; k0 += 4) {
      int kb = k0 + lhi * 2;
      v2f a;
      a.x = Hs[kb * HSTR + lnm];
      a.y = Hs[(kb + 1) * HSTR + lnm];
      size_t krow = (size_t)(half * 2048 + kb) * Dm;
      #pragma unroll
      for (int j = 0; j < 8; ++j) {
        int n = (wave * 8 + j) * 16 + lnm;
        v2f b;
        b.x = W2e[krow + n];
        b.y = W2e[krow + Dm + n];
        acc2[j] = __builtin_amdgcn_wmma_f32_16x16x4_f32(
            false, a, false, b, (short)0, acc2[j], false, false);
      }
    }
    __syncthreads();   // Hs reused by next half's FC1
  }

  // ---- bias + gate weight, scatter to per-(token, slot) contributions ----
  for (int j = 0; j < 8; ++j) {
    int   n  = (wave * 8 + j) * 16 + lnm;
    float bv = b2[(size_t)e * Dm + n];
    #pragma unroll
    for (int r = 0; r < 8; ++r) {
      int m = r + moff;
      if (m < rows) {
        float val = (acc2[j][r] + bv) * gl_w[m];
        contrib[((size_t)gl_g[m] * 4 + gl_k[m]) * Dm + n] = val;
      }
    }
  }
}

// ---------------------------------------------------------------------------
// 4) Deterministic reduction: fused[t] = sum over {view, k} of contributions.
// ---------------------------------------------------------------------------
__global__ void reduce_kernel(const float* __restrict__ contrib,
                              float* __restrict__ out) {
  size_t i = (size_t)blockIdx.x * blockDim.x + threadIdx.x;
  if (i >= (size_t)Nm * Dm) return;
  size_t t = i / Dm;
  size_t d = i - t * Dm;
  float s = 0.f;
  #pragma unroll
  for (int v = 0; v < 2; ++v)
    #pragma unroll
    for (int k = 0; k < 4; ++k)
      s += contrib[(((size_t)(v * Nm) + t) * 4 + k) * Dm + d];
  out[i] = s;
}

// ---------------------------------------------------------------------------
// Host launcher
// ---------------------------------------------------------------------------
extern "C" void kernel_launch(void* const* d_in, const int* in_sizes, int n_in,
                              void* d_out, int out_size, void* d_ws, size_t ws_size,
                              hipStream_t stream) {
  const float* v0   = (const float*)d_in[0];
  const float* v1   = (const float*)d_in[1];
  const float* W1   = (const float*)d_in[2];
  const float* b1   = (const float*)d_in[3];
  const float* W2   = (const float*)d_in[4];
  const float* b2   = (const float*)d_in[5];
  const float* rw0  = (const float*)d_in[6];
  const float* rb0  = (const float*)d_in[7];
  const float* rw1  = (const float*)d_in[8];
  const float* rb1  = (const float*)d_in[9];
  const float* keys = (const float*)d_in[10];
  float* out = (float*)d_out;

  char* ws = (char*)d_ws;
  int*   gidx    = (int*)(ws);                       //  64 KB
  float* gwt     = (float*)(ws + (64 << 10));        //  64 KB
  int*   counts  = (int*)(ws + (128 << 10));         //  64 B
  int*   lists   = (int*)(ws + (128 << 10) + 256);   // 256 KB
  float* contrib = (float*)(ws + (4 << 20));         //  64 MB

  gate_kernel<<<GTOT, 256, 0, stream>>>(v0, v1, keys, rw0, rb0, rw1, rb1,
                                        gidx, gwt);
  compact_kernel<<<Em, 32, 0, stream>>>(gidx, lists, counts);

  size_t ldsBytes = (size_t)(1024 * 16 + 2048 * HSTR) * sizeof(float); // 208 KB
  expert_kernel<<<dim3(CAP / 16, Em), 256, ldsBytes, stream>>>(
      v0, v1, W1, b1, W2, b2, lists, counts, gwt, contrib);

  size_t nOut = (size_t)Nm * Dm;
  reduce_kernel<<<(unsigned)((nOut + 255) / 256), 256, 0, stream>>>(contrib, out);
}